// DViLBlock_82858509075123
// MI455X (gfx1250) — compile-verified
//
#include <hip/hip_runtime.h>
#include <math.h>

typedef float v2f __attribute__((ext_vector_type(2)));
typedef float v8f __attribute__((ext_vector_type(8)));

#define DIM   192
#define INNER 384
#define NH    96
#define DH    4
#define HS    24
#define WS    24
#define SEQ   576
#define EPSN  1e-6f

// ---------------------------------------------------------------- LayerNorm
__global__ void ln_kernel(const float* __restrict__ x,
                          const float* __restrict__ w,
                          const float* __restrict__ b,
                          float* __restrict__ xn) {
    int row = blockIdx.x;          // 576 rows
    int tid = threadIdx.x;         // 192 threads
    __shared__ float red[DIM];
    float v = x[row * DIM + tid];
    red[tid] = v;
    __syncthreads();
    for (int st = 128; st > 0; st >>= 1) {
        if (tid < st && tid + st < DIM) red[tid] += red[tid + st];
        __syncthreads();
    }
    float mu = red[0] * (1.f / DIM);
    __syncthreads();
    float dv = v - mu;
    red[tid] = dv * dv;
    __syncthreads();
    for (int st = 128; st > 0; st >>= 1) {
        if (tid < st && tid + st < DIM) red[tid] += red[tid + st];
        __syncthreads();
    }
    float var = red[0] * (1.f / DIM);
    float rstd = rsqrtf(var + 1e-5f);
    xn[row * DIM + tid] = dv * rstd * w[tid] + b[tid];
}

// ------------------------------------------------- up GEMM (xn @ up_w^T + b)
// one wave per 16x16 tile; K=192 in steps of 4 via V_WMMA_F32_16X16X4_F32
__global__ void up_gemm_kernel(const float* __restrict__ A,   // [576,192]
                               const float* __restrict__ W,   // [768,192]
                               const float* __restrict__ bias,// [768]
                               float* __restrict__ xm,        // [576,384]
                               float* __restrict__ zout) {    // [576,384]
    int m0 = blockIdx.x * 16;
    int n0 = blockIdx.y * 16;
    int lane = threadIdx.x;
    int mr   = lane & 15;
    int half = lane >> 4;
    v8f c = {0.f,0.f,0.f,0.f,0.f,0.f,0.f,0.f};
    const float* arow = A + (m0 + mr) * DIM;
    const float* wrow = W + (n0 + mr) * DIM;
    for (int k0 = 0; k0 < DIM; k0 += 4) {
        int kk = k0 + 2 * half;
        v2f a, b;
        a.x = arow[kk];  a.y = arow[kk + 1];
        b.x = wrow[kk];  b.y = wrow[kk + 1];
        c = __builtin_amdgcn_wmma_f32_16x16x4_f32(false, a, false, b,
                                                  (short)0, c, false, false);
    }
    int col = n0 + mr;
    float bv = bias[col];
    for (int r = 0; r < 8; ++r) {
        int row = m0 + r + 8 * half;
        float val = c[r] + bv;
        if (col < INNER) xm[row * INNER + col]           = val;
        else             zout[row * INNER + (col-INNER)] = val;
    }
}

// -------------------------------------- depthwise 3x3 conv (SAME) + SiLU
__global__ void conv_silu_kernel(const float* __restrict__ xm,
                                 const float* __restrict__ cw, // [384,3,3]
                                 const float* __restrict__ cb,
                                 float* __restrict__ xa) {
    int idx = blockIdx.x * blockDim.x + threadIdx.x;   // 576*384
    int s = idx / INNER, c = idx % INNER;
    int h = s / WS, w = s % WS;
    float acc = cb[c];
    #pragma unroll
    for (int kh = 0; kh < 3; ++kh) {
        int ih = h + kh - 1;
        if (ih < 0 || ih >= HS) continue;
        #pragma unroll
        for (int kw = 0; kw < 3; ++kw) {
            int iw = w + kw - 1;
            if (iw < 0 || iw >= WS) continue;
            acc += xm[(ih * WS + iw) * INNER + c] * cw[c * 9 + kh * 3 + kw];
        }
    }
    float sig = 1.f / (1.f + __expf(-acc));
    xa[idx] = acc * sig;
}

// ---------------------------------------------- headwise q/k/v (4x4 blocks)
__global__ void qkv_kernel(const float* __restrict__ xa,
                           const float* __restrict__ xm,
                           const float* __restrict__ qw, const float* __restrict__ qb_,
                           const float* __restrict__ kw, const float* __restrict__ kb_,
                           const float* __restrict__ vw, const float* __restrict__ vb_,
                           float* __restrict__ q, float* __restrict__ k,
                           float* __restrict__ v) {
    int idx = blockIdx.x * blockDim.x + threadIdx.x;   // 576*384
    int s = idx / INNER, c = idx % INNER;
    int n = c >> 2, o = c & 3;
    const float* xq = xa + s * INNER + n * 4;
    const float* xv = xm + s * INNER + n * 4;
    float aq = qb_[c], ak = kb_[c], av = vb_[c];
    #pragma unroll
    for (int i = 0; i < 4; ++i) {
        float xi = xq[i];
        aq += xi * qw[n * 16 + o * 4 + i];
        ak += xi * kw[n * 16 + o * 4 + i];
        av += xv[i] * vw[n * 16 + o * 4 + i];
    }
    q[idx] = aq; k[idx] = ak; v[idx] = av;
}

// ------------------------------------- gates: ig/fg = gin @ W^T + b, [NH,S]
__global__ void gates_kernel(const float* __restrict__ q,
                             const float* __restrict__ k,
                             const float* __restrict__ v,
                             const float* __restrict__ igw,
                             const float* __restrict__ igb_,
                             const float* __restrict__ fgw,
                             const float* __restrict__ fgb_,
                             float* __restrict__ ig, float* __restrict__ fg) {
    int wv   = (blockIdx.x * blockDim.x + threadIdx.x) >> 5; // wave id, 96*576
    int lane = threadIdx.x & 31;
    int head = wv / SEQ, s = wv % SEQ;
    float ai = 0.f, af = 0.f;
    for (int j = lane; j < 3 * INNER; j += 32) {
        float g = (j < INNER)     ? q[s * INNER + j]
                : (j < 2 * INNER) ? k[s * INNER + j - INNER]
                                  : v[s * INNER + j - 2 * INNER];
        ai += g * igw[head * 3 * INNER + j];
        af += g * fgw[head * 3 * INNER + j];
    }
    for (int m = 1; m <= 16; m <<= 1) {
        ai += __shfl_xor(ai, m, 32);
        af += __shfl_xor(af, m, 32);
    }
    if (lane == 0) {
        ig[head * SEQ + s] = ai + igb_[head];
        fg[head * SEQ + s] = af + fgb_[head];
    }
}

// --------------- per-head scan: lfc = cumsum(logsigmoid(fg)); maxD prefix
__global__ void scan_kernel(const float* __restrict__ ig,
                            const float* __restrict__ fg,
                            float* __restrict__ lfc, float* __restrict__ md) {
    int head = blockIdx.x * blockDim.x + threadIdx.x;
    if (head >= NH) return;
    float cum = 0.f, m = -3.4e38f;
    for (int s = 0; s < SEQ; ++s) {
        float f = fg[head * SEQ + s];
        float ls = (f > 0.f) ? -log1pf(__expf(-f)) : (f - log1pf(__expf(f)));
        cum += ls;
        lfc[head * SEQ + s] = cum;
        float cand = ig[head * SEQ + s] - cum;
        m = fmaxf(m, cand);
        md[head * SEQ + s] = cum + m;   // maxD[s] = lfc[s] + max_{t<=s}(ig[t]-lfc[t])
    }
}

// ---------- attention: one wave per (head, 16-row tile); K=DH=4 -> 1 WMMA/tile
// fused epilogue: norm divide, per-head groupnorm, +skip*xa, *silu(z) -> hh
__global__ void attn_kernel(const float* __restrict__ q,
                            const float* __restrict__ k,
                            const float* __restrict__ v,
                            const float* __restrict__ ig,
                            const float* __restrict__ lfc,
                            const float* __restrict__ md,
                            const float* __restrict__ onw,
                            const float* __restrict__ skip,
                            const float* __restrict__ xa,
                            const float* __restrict__ z,
                            float* __restrict__ hh) {
    int head = blockIdx.x;          // 96
    int s0   = blockIdx.y * 16;     // 36 tiles
    int lane = threadIdx.x;
    int nn   = lane & 15;
    int half = lane >> 4;
    __shared__ float sh_lfc[16], sh_md[16];
    if (lane < 16) {
        sh_lfc[lane] = lfc[head * SEQ + s0 + lane];
        sh_md[lane]  = md [head * SEQ + s0 + lane];
    }
    __syncthreads();

    int kk = 2 * half;
    v2f a;
    a.x = q[(s0 + nn) * INNER + head * 4 + kk];
    a.y = q[(s0 + nn) * INNER + head * 4 + kk + 1];

    float rowsum[8];
    float hacc[8][4];
    #pragma unroll
    for (int r = 0; r < 8; ++r) {
        rowsum[r] = 0.f;
        hacc[r][0] = hacc[r][1] = hacc[r][2] = hacc[r][3] = 0.f;
    }

    for (int t0 = 0; t0 <= s0; t0 += 16) {
        v2f b;
        b.x = k[(t0 + nn) * INNER + head * 4 + kk];
        b.y = k[(t0 + nn) * INNER + head * 4 + kk + 1];
        v8f c = {0.f,0.f,0.f,0.f,0.f,0.f,0.f,0.f};
        c = __builtin_amdgcn_wmma_f32_16x16x4_f32(false, a, false, b,
                                                  (short)0, c, false, false);
        int   t    = t0 + nn;                     // this lane's score column
        float lfct = lfc[head * SEQ + t];
        float igt  = ig [head * SEQ + t];
        float v0 = v[t * INNER + head * 4 + 0];
        float v1 = v[t * INNER + head * 4 + 1];
        float v2 = v[t * INNER + head * 4 + 2];
        float v3 = v[t * INNER + head * 4 + 3];
        #pragma unroll
        for (int r = 0; r < 8; ++r) {
            int M = r + 8 * half;
            int s = s0 + M;
            float cv = 0.f;
            if (t <= s) {
                float wgt = __expf(sh_lfc[M] - lfct + igt - sh_md[M]);
                cv = c[r] * 0.5f * wgt;           // 1/sqrt(DH)=0.5
            }
            rowsum[r] += cv;
            hacc[r][0] += cv * v0;  hacc[r][1] += cv * v1;
            hacc[r][2] += cv * v2;  hacc[r][3] += cv * v3;
        }
    }

    // reduce across the 16 lanes holding the same row block
    for (int m = 1; m <= 8; m <<= 1) {
        #pragma unroll
        for (int r = 0; r < 8; ++r) {
            rowsum[r]  += __shfl_xor(rowsum[r],  m, 32);
            hacc[r][0] += __shfl_xor(hacc[r][0], m, 32);
            hacc[r][1] += __shfl_xor(hacc[r][1], m, 32);
            hacc[r][2] += __shfl_xor(hacc[r][2], m, 32);
            hacc[r][3] += __shfl_xor(hacc[r][3], m, 32);
        }
    }

    if (nn == 0) {
        for (int r = 0; r < 8; ++r) {
            int M = r + 8 * half;
            int s = s0 + M;
            float norm = fmaxf(fabsf(rowsum[r]), __expf(-sh_md[M]));
            float inv  = 1.f / (norm + EPSN);
            float hv[4], mu = 0.f;
            #pragma unroll
            for (int d = 0; d < 4; ++d) { hv[d] = hacc[r][d] * inv; mu += hv[d]; }
            mu *= 0.25f;
            float var = 0.f;
            #pragma unroll
            for (int d = 0; d < 4; ++d) { float dd = hv[d] - mu; var += dd * dd; }
            var *= 0.25f;
            float rstd = rsqrtf(var + 1e-5f);
            #pragma unroll
            for (int d = 0; d < 4; ++d) {
                int  cidx = head * 4 + d;
                float hn  = (hv[d] - mu) * rstd * onw[cidx];
                float zv  = z[s * INNER + cidx];
                float sig = 1.f / (1.f + __expf(-zv));
                hh[s * INNER + cidx] =
                    (hn + skip[cidx] * xa[s * INNER + cidx]) * (zv * sig);
            }
        }
    }
}

// -------------------------------------- down GEMM: out = hh @ down_w^T + b
__global__ void down_gemm_kernel(const float* __restrict__ A,   // [576,384]
                                 const float* __restrict__ W,   // [192,384]
                                 const float* __restrict__ bias,// [192]
                                 float* __restrict__ out) {     // [576,192]
    int m0 = blockIdx.x * 16;
    int n0 = blockIdx.y * 16;
    int lane = threadIdx.x;
    int mr   = lane & 15;
    int half = lane >> 4;
    v8f c = {0.f,0.f,0.f,0.f,0.f,0.f,0.f,0.f};
    const float* arow = A + (m0 + mr) * INNER;
    const float* wrow = W + (n0 + mr) * INNER;
    for (int k0 = 0; k0 < INNER; k0 += 4) {
        int kk = k0 + 2 * half;
        v2f a, b;
        a.x = arow[kk];  a.y = arow[kk + 1];
        b.x = wrow[kk];  b.y = wrow[kk + 1];
        c = __builtin_amdgcn_wmma_f32_16x16x4_f32(false, a, false, b,
                                                  (short)0, c, false, false);
    }
    int col = n0 + mr;
    float bv = bias[col];
    for (int r = 0; r < 8; ++r)
        out[(m0 + r + 8 * half) * DIM + col] = c[r] + bv;
}

// ----------------------------------------------------------------- launcher
extern "C" void kernel_launch(void* const* d_in, const int* in_sizes, int n_in,
                              void* d_out, int out_size, void* d_ws, size_t ws_size,
                              hipStream_t stream) {
    const float* x      = (const float*)d_in[0];
    // d_in[1] (d) feeds only dead code -> skipped
    const float* ln_w   = (const float*)d_in[2];
    const float* ln_b   = (const float*)d_in[3];
    const float* up_w   = (const float*)d_in[4];
    const float* up_b   = (const float*)d_in[5];
    const float* q_w    = (const float*)d_in[8];
    const float* q_b    = (const float*)d_in[9];
    const float* k_w    = (const float*)d_in[10];
    const float* k_b    = (const float*)d_in[11];
    const float* v_w    = (const float*)d_in[12];
    const float* v_b    = (const float*)d_in[13];
    const float* conv_w = (const float*)d_in[14];
    const float* conv_b = (const float*)d_in[15];
    const float* ig_w   = (const float*)d_in[16];
    const float* ig_b   = (const float*)d_in[17];
    const float* fg_w   = (const float*)d_in[18];
    const float* fg_b   = (const float*)d_in[19];
    const float* onw    = (const float*)d_in[20];
    const float* skip   = (const float*)d_in[21];
    const float* down_w = (const float*)d_in[22];
    const float* down_b = (const float*)d_in[23];
    float* out = (float*)d_out;

    float* w = (float*)d_ws;
    float* xn  = w;                 // 576*192
    float* xm  = xn  + SEQ * DIM;   // 576*384
    float* z   = xm  + SEQ * INNER;
    float* xa  = z   + SEQ * INNER;
    float* qb  = xa  + SEQ * INNER;
    float* kb  = qb  + SEQ * INNER;
    float* vb  = kb  + SEQ * INNER;
    float* hh  = vb  + SEQ * INNER;
    float* igB = hh  + SEQ * INNER; // 96*576
    float* fgB = igB + NH * SEQ;
    float* lfc = fgB + NH * SEQ;
    float* md  = lfc + NH * SEQ;

    ln_kernel<<<SEQ, DIM, 0, stream>>>(x, ln_w, ln_b, xn);
    up_gemm_kernel<<<dim3(SEQ / 16, (2 * INNER) / 16), 32, 0, stream>>>(
        xn, up_w, up_b, xm, z);
    conv_silu_kernel<<<(SEQ * INNER) / 256, 256, 0, stream>>>(xm, conv_w, conv_b, xa);
    qkv_kernel<<<(SEQ * INNER) / 256, 256, 0, stream>>>(
        xa, xm, q_w, q_b, k_w, k_b, v_w, v_b, qb, kb, vb);
    gates_kernel<<<(NH * SEQ * 32) / 256, 256, 0, stream>>>(
        qb, kb, vb, ig_w, ig_b, fg_w, fg_b, igB, fgB);
    scan_kernel<<<1, 128, 0, stream>>>(igB, fgB, lfc, md);
    attn_kernel<<<dim3(NH, SEQ / 16), 32, 0, stream>>>(
        qb, kb, vb, igB, lfc, md, onw, skip, xa, z, hh);
    down_gemm_kernel<<<dim3(SEQ / 16, DIM / 16), 32, 0, stream>>>(
        hh, down_w, down_b, out);
    (void)in_sizes; (void)n_in; (void)out_size; (void)ws_size;
}